// gdnCell_70944269796144
// MI455X (gfx1250) — compile-verified
//
#include <hip/hip_runtime.h>

// ---------------------------------------------------------------------------
// gdnCell on MI455X (gfx1250): fp32 WMMA (V_WMMA_F32_16X16X4_F32) GEMMs +
// fused elementwise / reduction passes. wave32, 16x16 tiles, 2x2 tiles/wave.
// ---------------------------------------------------------------------------

typedef float v2f __attribute__((ext_vector_type(2)));
typedef float v8f __attribute__((ext_vector_type(8)));

#define NMAT 512
#define SLICE (NMAT * NMAT)   // 262144 elements per 512x512 matrix

// D = A(16x4) x B(4x16) + C, fp32 in/out, wave32.
__device__ __forceinline__ v8f wmma4(v2f a, v2f b, v8f c) {
  // 8 args: (neg_a, A, neg_b, B, c_mod, C, reuse_a, reuse_b)
  return __builtin_amdgcn_wmma_f32_16x16x4_f32(
      false, a, false, b, (short)0, c, false, false);
}

// A-fragment (16x4): lane l holds row m = mBase + (l&15),
// K pair k = kk + ((l>>4)<<1) -> {X[m][k], X[m][k+1]}
__device__ __forceinline__ v2f load_afrag(const float* __restrict__ X, int m, int k) {
  const float* p = X + m * NMAT + k;
  v2f r; r.x = p[0]; r.y = p[1];
  return r;
}

// B-fragment (4x16): lane l holds col n = nBase + (l&15),
// K pair k = kk + ((l>>4)<<1) -> {Y[k][n], Y[k+1][n]}
__device__ __forceinline__ v2f load_bfrag(const float* __restrict__ Y, int k, int n) {
  const float* p = Y + k * NMAT + n;
  v2f r; r.x = p[0]; r.y = p[NMAT];
  return r;
}

// acc(32x32) += X(512x512) @ Y(512x512) restricted to [mBase..+32) x [nBase..+32)
__device__ __forceinline__ void wave_gemm(const float* __restrict__ X,
                                          const float* __restrict__ Y,
                                          int mBase, int nBase, int lane,
                                          v8f acc[2][2]) {
  const int mr = lane & 15;
  const int nc = lane & 15;
  const int ko = (lane >> 4) << 1;   // 0 or 2
  for (int kk = 0; kk < NMAT; kk += 4) {
    const int k = kk + ko;
    v2f a0 = load_afrag(X, mBase + mr, k);
    v2f a1 = load_afrag(X, mBase + 16 + mr, k);
    v2f b0 = load_bfrag(Y, k, nBase + nc);
    v2f b1 = load_bfrag(Y, k, nBase + 16 + nc);
    acc[0][0] = wmma4(a0, b0, acc[0][0]);
    acc[0][1] = wmma4(a0, b1, acc[0][1]);
    acc[1][0] = wmma4(a1, b0, acc[1][0]);
    acc[1][1] = wmma4(a1, b1, acc[1][1]);
  }
}

// acc(32x32) += X @ Y + Y @ X  (for mix = s@A + A@s), same tile.
__device__ __forceinline__ void wave_gemm_mix(const float* __restrict__ X,
                                              const float* __restrict__ Y,
                                              int mBase, int nBase, int lane,
                                              v8f acc[2][2]) {
  const int mr = lane & 15;
  const int nc = lane & 15;
  const int ko = (lane >> 4) << 1;
  for (int kk = 0; kk < NMAT; kk += 4) {
    const int k = kk + ko;
    // X rows / Y cols  (X @ Y)
    v2f ax0 = load_afrag(X, mBase + mr, k);
    v2f ax1 = load_afrag(X, mBase + 16 + mr, k);
    v2f by0 = load_bfrag(Y, k, nBase + nc);
    v2f by1 = load_bfrag(Y, k, nBase + 16 + nc);
    // Y rows / X cols  (Y @ X)
    v2f ay0 = load_afrag(Y, mBase + mr, k);
    v2f ay1 = load_afrag(Y, mBase + 16 + mr, k);
    v2f bx0 = load_bfrag(X, k, nBase + nc);
    v2f bx1 = load_bfrag(X, k, nBase + 16 + nc);
    acc[0][0] = wmma4(ax0, by0, acc[0][0]);
    acc[0][1] = wmma4(ax0, by1, acc[0][1]);
    acc[1][0] = wmma4(ax1, by0, acc[1][0]);
    acc[1][1] = wmma4(ax1, by1, acc[1][1]);
    acc[0][0] = wmma4(ay0, bx0, acc[0][0]);
    acc[0][1] = wmma4(ay0, bx1, acc[0][1]);
    acc[1][0] = wmma4(ay1, bx0, acc[1][0]);
    acc[1][1] = wmma4(ay1, bx1, acc[1][1]);
  }
}

// Block: 128 threads = 4 waves in 2x2 -> 64x64 output tile per block.
// grid = (8, 8, nInstances)

// T2[b] = 2 * A[b] @ A[b] - I
__global__ __launch_bounds__(128) void cheb1_kernel(const float* __restrict__ A,
                                                    float* __restrict__ T2) {
  const int b = blockIdx.z;
  const float* Ab = A + b * SLICE;
  float* D = T2 + b * SLICE;
  const int lane = threadIdx.x & 31;
  const int w = threadIdx.x >> 5;
  const int mBase = blockIdx.y * 64 + (w >> 1) * 32;
  const int nBase = blockIdx.x * 64 + (w & 1) * 32;
  v8f acc[2][2] = {};
  wave_gemm(Ab, Ab, mBase, nBase, lane, acc);
  const int half8 = (lane >> 4) << 3;
  const int nl = lane & 15;
#pragma unroll
  for (int tr = 0; tr < 2; ++tr)
#pragma unroll
    for (int tc = 0; tc < 2; ++tc)
#pragma unroll
      for (int r = 0; r < 8; ++r) {
        const int m = mBase + tr * 16 + r + half8;
        const int n = nBase + tc * 16 + nl;
        D[m * NMAT + n] = 2.0f * acc[tr][tc][r] - ((m == n) ? 1.0f : 0.0f);
      }
}

// T3[b] = 2 * A[b] @ T2[b] - A[b]
__global__ __launch_bounds__(128) void cheb2_kernel(const float* __restrict__ A,
                                                    const float* __restrict__ T2,
                                                    float* __restrict__ T3) {
  const int b = blockIdx.z;
  const float* Ab = A + b * SLICE;
  const float* T2b = T2 + b * SLICE;
  float* D = T3 + b * SLICE;
  const int lane = threadIdx.x & 31;
  const int w = threadIdx.x >> 5;
  const int mBase = blockIdx.y * 64 + (w >> 1) * 32;
  const int nBase = blockIdx.x * 64 + (w & 1) * 32;
  v8f acc[2][2] = {};
  wave_gemm(Ab, T2b, mBase, nBase, lane, acc);
  const int half8 = (lane >> 4) << 3;
  const int nl = lane & 15;
#pragma unroll
  for (int tr = 0; tr < 2; ++tr)
#pragma unroll
    for (int tc = 0; tc < 2; ++tc)
#pragma unroll
      for (int r = 0; r < 8; ++r) {
        const int m = mBase + tr * 16 + r + half8;
        const int n = nBase + tc * 16 + nl;
        D[m * NMAT + n] = 2.0f * acc[tr][tc][r] - Ab[m * NMAT + n];
      }
}

// M[i,b] = s_in[i,b] @ A[b] + A[b] @ s_in[i,b] ; grid.z = i*8 + b (32)
__global__ __launch_bounds__(128) void mix_kernel(const float* __restrict__ s_in,
                                                  const float* __restrict__ a_o,
                                                  float* __restrict__ M) {
  const int z = blockIdx.z;            // i*8 + b
  const int b = z & 7;
  const float* Sb = s_in + z * SLICE;  // (i*8+b) slice of s_in
  const float* Ab = a_o + b * SLICE;
  float* D = M + z * SLICE;
  const int lane = threadIdx.x & 31;
  const int w = threadIdx.x >> 5;
  const int mBase = blockIdx.y * 64 + (w >> 1) * 32;
  const int nBase = blockIdx.x * 64 + (w & 1) * 32;
  v8f acc[2][2] = {};
  wave_gemm_mix(Sb, Ab, mBase, nBase, lane, acc);
  const int half8 = (lane >> 4) << 3;
  const int nl = lane & 15;
#pragma unroll
  for (int tr = 0; tr < 2; ++tr)
#pragma unroll
    for (int tc = 0; tc < 2; ++tc)
#pragma unroll
      for (int r = 0; r < 8; ++r) {
        const int m = mBase + tr * 16 + r + half8;
        const int n = nBase + tc * 16 + nl;
        D[m * NMAT + n] = acc[tr][tc][r];
      }
}

// out_pre[o,b] = 0.25 * ( sum_i alpha[o,i]*s_in[i,b] + sum_i beta[o,i]*M[i,b]
//                        + cbar[o,1]*A[b] + cbar[o,2]*T2[b] + cbar[o,3]*T3[b] )
// with diagonal zeroed (T0 term lands only on the diagonal -> dropped).
// grid = (SLICE/256, 8)
__global__ __launch_bounds__(256) void combine_kernel(
    const float* __restrict__ s_in, const float* __restrict__ a_o,
    const float* __restrict__ M, const float* __restrict__ T2,
    const float* __restrict__ T3, const float* __restrict__ alpha,
    const float* __restrict__ beta, const float* __restrict__ coeffs,
    float* __restrict__ out_pre) {
  const int b = blockIdx.y;
  const int nm = blockIdx.x * blockDim.x + threadIdx.x;  // 0..SLICE-1
  const int n = nm >> 9;
  const int m = nm & (NMAT - 1);
  const bool diag = (n == m);

  float sv[4], mv[4];
#pragma unroll
  for (int i = 0; i < 4; ++i) {
    sv[i] = s_in[(i * 8 + b) * SLICE + nm];
    mv[i] = M[(i * 8 + b) * SLICE + nm];
  }
  const float av = a_o[b * SLICE + nm];
  const float t2 = T2[b * SLICE + nm];
  const float t3 = T3[b * SLICE + nm];

#pragma unroll
  for (int o = 0; o < 8; ++o) {
    float acc = 0.0f;
    float c1 = 0.0f, c2 = 0.0f, c3 = 0.0f;
#pragma unroll
    for (int i = 0; i < 4; ++i) {
      acc += alpha[o * 4 + i] * sv[i] + beta[o * 4 + i] * mv[i];
      c1 += coeffs[o * 16 + 4 + i];    // Cheb T1 coeff, summed over c_in
      c2 += coeffs[o * 16 + 8 + i];    // T2
      c3 += coeffs[o * 16 + 12 + i];   // T3
    }
    acc += c1 * av + c2 * t2 + c3 * t3;
    out_pre[(o * 8 + b) * SLICE + nm] = diag ? 0.0f : 0.25f * acc;
  }
}

// Per (o,b) slice max-abs. grid = 64, block = 256.
__global__ __launch_bounds__(256) void maxabs_kernel(const float* __restrict__ out_pre,
                                                     float* __restrict__ mx) {
  const int slice = blockIdx.x;  // o*8 + b
  const float* p = out_pre + (size_t)slice * SLICE;
  float v = 0.0f;
  for (int idx = threadIdx.x; idx < SLICE; idx += 256)
    v = fmaxf(v, fabsf(p[idx]));
  __shared__ float red[256];
  red[threadIdx.x] = v;
  __syncthreads();
  for (int s = 128; s > 0; s >>= 1) {
    if (threadIdx.x < s)
      red[threadIdx.x] = fmaxf(red[threadIdx.x], red[threadIdx.x + s]);
    __syncthreads();
  }
  if (threadIdx.x == 0) mx[slice] = (red[0] == 0.0f) ? 1.0f : red[0];
}

// out = relu(out/mx[slice] - tau[o]), in place. grid = 65536, block = 256.
__global__ __launch_bounds__(256) void finalize_kernel(float* __restrict__ out,
                                                       const float* __restrict__ mx,
                                                       const float* __restrict__ tau) {
  const int idx = blockIdx.x * 256 + threadIdx.x;
  const int slice = idx >> 18;  // / SLICE
  const int o = slice >> 3;
  const float v = out[idx] / mx[slice] - tau[o];
  out[idx] = v > 0.0f ? v : 0.0f;
}

extern "C" void kernel_launch(void* const* d_in, const int* in_sizes, int n_in,
                              void* d_out, int out_size, void* d_ws, size_t ws_size,
                              hipStream_t stream) {
  (void)in_sizes; (void)n_in; (void)out_size; (void)ws_size;
  const float* s_in   = (const float*)d_in[0];  // (4,8,512,512)
  const float* a_o    = (const float*)d_in[1];  // (8,512,512)
  const float* alpha  = (const float*)d_in[2];  // (8,4)
  const float* beta   = (const float*)d_in[3];  // (8,4)
  const float* coeffs = (const float*)d_in[4];  // (8,4,4)
  const float* tau    = (const float*)d_in[5];  // (8)
  float* out = (float*)d_out;                   // (8,8,512,512)

  char* ws = (char*)d_ws;
  float* M  = (float*)(ws);                       // 32 * SLICE floats = 32 MB
  float* T2 = (float*)(ws + 33554432ull);         //  8 * SLICE floats =  8 MB
  float* T3 = (float*)(ws + 41943040ull);         //  8 * SLICE floats =  8 MB
  float* mx = (float*)(ws + 50331648ull);         // 64 floats

  const dim3 gblk(128);
  cheb1_kernel<<<dim3(8, 8, 8),  gblk, 0, stream>>>(a_o, T2);
  cheb2_kernel<<<dim3(8, 8, 8),  gblk, 0, stream>>>(a_o, T2, T3);
  mix_kernel  <<<dim3(8, 8, 32), gblk, 0, stream>>>(s_in, a_o, M);
  combine_kernel<<<dim3(SLICE / 256, 8), 256, 0, stream>>>(
      s_in, a_o, M, T2, T3, alpha, beta, coeffs, out);
  maxabs_kernel<<<64, 256, 0, stream>>>(out, mx);
  finalize_kernel<<<(8 * 8 * SLICE) / 256, 256, 0, stream>>>(out, mx, tau);
}